// GAT_73933567033778
// MI455X (gfx1250) — compile-verified
//
#include <hip/hip_runtime.h>
#include <math.h>

// ---------------------------------------------------------------------------
// 2-layer GAT for gfx1250 (MI455X).
//  - GEMMs via V_WMMA_F32_16X16X4_F32 (fp32 WMMA, keeps reference numerics)
//  - edge softmax via 3 passes (ordered-uint atomic max, exp+atomic sum,
//    normalized atomic aggregation)
// ---------------------------------------------------------------------------

typedef float v2f __attribute__((ext_vector_type(2)));
typedef float v8f __attribute__((ext_vector_type(8)));

// ---------------- fp32 WMMA GEMM: C[M,N] = A[M,K] * B[K,N] ------------------
// One wave (32 lanes) per block computes one 16x16 output tile.
// M, N, K must be multiples of 16 (10000, 256/64, 256 all are).
__global__ void gat_gemm_wmma_f32(const float* __restrict__ A,
                                  const float* __restrict__ B,
                                  float* __restrict__ C,
                                  int M, int N, int K) {
  const int tile_m = blockIdx.x << 4;
  const int tile_n = blockIdx.y << 4;
  const int lane   = threadIdx.x;        // 0..31
  const int r      = lane & 15;          // row (A) / col (B) within tile
  const int kh     = (lane >> 4) << 1;   // K sub-offset: 0 (lanes 0-15) or 2

  v8f acc = {};
  const float* Arow = A + (size_t)(tile_m + r) * K;
#pragma unroll 4
  for (int k = 0; k < K; k += 4) {
    // A 16x4 fragment (ISA 7.12.2: lanes 0-15 -> K=0,1 ; lanes 16-31 -> K=2,3)
    v2f a, b;
    a.x = Arow[k + kh];
    a.y = Arow[k + kh + 1];
    // B 4x16 fragment: row K striped across lanes, symmetric K split
    b.x = B[(size_t)(k + kh) * N + tile_n + r];
    b.y = B[(size_t)(k + kh + 1) * N + tile_n + r];
    acc = __builtin_amdgcn_wmma_f32_16x16x4_f32(
        /*neg_a=*/false, a, /*neg_b=*/false, b,
        /*c_mod=*/(short)0, acc, /*reuse_a=*/false, /*reuse_b=*/false);
  }

  // C/D layout: VGPR i -> M = i + 8*(lane>=16), N = lane&15
  const int mbase = tile_m + ((lane >> 4) << 3);
#pragma unroll
  for (int i = 0; i < 8; ++i)
    C[(size_t)(mbase + i) * N + tile_n + (lane & 15)] = acc[i];
}

// ---------------- per-node attention scores a_s, a_d ------------------------
__global__ void gat_att_scores(const float* __restrict__ H,
                               const float* __restrict__ att_s,
                               const float* __restrict__ att_d,
                               float* __restrict__ as_, float* __restrict__ ad_,
                               int N, int Hh, int C) {
  int i = blockIdx.x * blockDim.x + threadIdx.x;   // over N*Hh
  if (i >= N * Hh) return;
  int n = i / Hh, h = i % Hh;
  const float* hp = H + ((size_t)n * Hh + h) * C;
  const float* ws = att_s + (size_t)h * C;
  const float* wd = att_d + (size_t)h * C;
  float s = 0.f, d = 0.f;
  for (int c = 0; c < C; ++c) { float v = hp[c]; s += v * ws[c]; d += v * wd[c]; }
  as_[i] = s; ad_[i] = d;
}

// ---------------- ordered-uint encoding for float atomic max ----------------
__device__ __forceinline__ unsigned f32_ord(float f) {
  unsigned u = __float_as_uint(f);
  return (u & 0x80000000u) ? ~u : (u | 0x80000000u);
}
__device__ __forceinline__ float ord_f32(unsigned k) {
  unsigned u = (k & 0x80000000u) ? (k & 0x7FFFFFFFu) : ~k;
  return __uint_as_float(u);
}

__device__ __forceinline__ void edge_nodes(const long long* __restrict__ ei,
                                           int E, int e, int& s, int& d) {
  if (e < E) { s = (int)ei[e]; d = (int)ei[E + e]; }
  else       { s = d = e - E; }            // self loop
}

__device__ __forceinline__ float leaky02(float v) {
  return v > 0.f ? v : 0.2f * v;
}

// ---------------- pass 1: segment max (atomic on ordered uint) --------------
__global__ void gat_edge_max(const long long* __restrict__ ei, int E, int nE,
                             const float* __restrict__ as_,
                             const float* __restrict__ ad_,
                             unsigned* __restrict__ mkey, int Hh) {
  int i = blockIdx.x * blockDim.x + threadIdx.x;   // over nE*Hh
  if (i >= nE * Hh) return;
  int e = i / Hh, h = i % Hh, s, d;
  edge_nodes(ei, E, e, s, d);
  float v = leaky02(as_[(size_t)s * Hh + h] + ad_[(size_t)d * Hh + h]);
  atomicMax(&mkey[(size_t)d * Hh + h], f32_ord(v));
}

// ---------------- pass 2: exp(e - m), store weight, segment sum -------------
__global__ void gat_edge_sum(const long long* __restrict__ ei, int E, int nE,
                             const float* __restrict__ as_,
                             const float* __restrict__ ad_,
                             const unsigned* __restrict__ mkey,
                             float* __restrict__ alpha,
                             float* __restrict__ ssum, int Hh) {
  int i = blockIdx.x * blockDim.x + threadIdx.x;
  if (i >= nE * Hh) return;
  int e = i / Hh, h = i % Hh, s, d;
  edge_nodes(ei, E, e, s, d);
  float v  = leaky02(as_[(size_t)s * Hh + h] + ad_[(size_t)d * Hh + h]);
  float ex = expf(v - ord_f32(mkey[(size_t)d * Hh + h]));
  alpha[i] = ex;
  atomicAdd(&ssum[(size_t)d * Hh + h], ex);
}

// ---------------- pass 3: normalized message aggregation --------------------
// One block per edge; blockDim = Hh*C threads (256 for layer1, 64 for layer2)
// so the h[src] row read is a single coalesced burst.
__global__ void gat_edge_agg(const long long* __restrict__ ei, int E,
                             const float* __restrict__ Hfeat,
                             const float* __restrict__ alpha,
                             const float* __restrict__ ssum,
                             float* __restrict__ out, int Hh, int C) {
  int e = blockIdx.x;
  int t = threadIdx.x;
  int h = t / C, c = t % C;
  int s, d;
  edge_nodes(ei, E, e, s, d);
  float a = alpha[(size_t)e * Hh + h] / (ssum[(size_t)d * Hh + h] + 1e-16f);
  float m = Hfeat[((size_t)s * Hh + h) * C + c] * a;
  atomicAdd(&out[((size_t)d * Hh + h) * C + c], m);
}

// ---------------- small utility kernels -------------------------------------
__global__ void gat_fill_u32(unsigned* __restrict__ p, unsigned v, long long n) {
  long long i = (long long)blockIdx.x * blockDim.x + threadIdx.x;
  if (i < n) p[i] = v;
}

__global__ void gat_init_bias(float* __restrict__ o, const float* __restrict__ b,
                              int D, long long n) {
  long long i = (long long)blockIdx.x * blockDim.x + threadIdx.x;
  if (i < n) o[i] = b[i % D];
}

__global__ void gat_bias_relu(float* __restrict__ o, const float* __restrict__ b,
                              int D, long long n) {
  long long i = (long long)blockIdx.x * blockDim.x + threadIdx.x;
  if (i < n) { float v = o[i] + b[i % D]; o[i] = v > 0.f ? v : 0.f; }
}

// ---------------------------------------------------------------------------
extern "C" void kernel_launch(void* const* d_in, const int* in_sizes, int n_in,
                              void* d_out, int out_size, void* d_ws, size_t ws_size,
                              hipStream_t stream) {
  const float*     x    = (const float*)d_in[0];
  const long long* ei   = (const long long*)d_in[1];   // int64 per reference
  const float*     W1   = (const float*)d_in[2];
  const float*     aS1  = (const float*)d_in[3];
  const float*     aD1  = (const float*)d_in[4];
  const float*     b1   = (const float*)d_in[5];
  const float*     W2   = (const float*)d_in[6];
  const float*     aS2  = (const float*)d_in[7];
  const float*     aD2  = (const float*)d_in[8];
  const float*     b2   = (const float*)d_in[9];
  float*           out  = (float*)d_out;

  const int NF = 256, HID = 256, H1h = 8, C1 = 32, C2 = 64;
  const int N  = in_sizes[0] / NF;
  const int E  = in_sizes[1] / 2;
  const int nE = E + N;

  // ---- workspace sub-allocation (256B aligned) ----
  char* wp = (char*)d_ws;
  auto alloc = [&](size_t bytes) -> char* {
    char* p = wp; wp += (bytes + 255) & ~(size_t)255; return p;
  };
  float*    H1  = (float*)   alloc((size_t)N * HID * 4);
  float*    O1  = (float*)   alloc((size_t)N * HID * 4);
  float*    as1 = (float*)   alloc((size_t)N * H1h * 4);
  float*    ad1 = (float*)   alloc((size_t)N * H1h * 4);
  unsigned* m1  = (unsigned*)alloc((size_t)N * H1h * 4);
  float*    s1  = (float*)   alloc((size_t)N * H1h * 4);
  float*    al1 = (float*)   alloc((size_t)nE * H1h * 4);
  float*    H2  = (float*)   alloc((size_t)N * C2 * 4);
  float*    as2 = (float*)   alloc((size_t)N * 4);
  float*    ad2 = (float*)   alloc((size_t)N * 4);
  unsigned* m2  = (unsigned*)alloc((size_t)N * 4);
  float*    s2  = (float*)   alloc((size_t)N * 4);
  float*    al2 = (float*)   alloc((size_t)nE * 4);
  (void)ws_size; (void)n_in; (void)out_size;

  const int TB = 256;
  auto cdiv = [](long long a, long long b) { return (int)((a + b - 1) / b); };

  // ================= Layer 1 =================
  {
    dim3 g(N / 16, HID / 16);
    gat_gemm_wmma_f32<<<g, 32, 0, stream>>>(x, W1, H1, N, HID, NF);
  }
  gat_att_scores<<<cdiv((long long)N * H1h, TB), TB, 0, stream>>>(
      H1, aS1, aD1, as1, ad1, N, H1h, C1);

  gat_fill_u32<<<cdiv((long long)N * H1h, TB), TB, 0, stream>>>(m1, 0u, (long long)N * H1h);
  gat_fill_u32<<<cdiv((long long)N * H1h, TB), TB, 0, stream>>>((unsigned*)s1, 0u, (long long)N * H1h);
  gat_fill_u32<<<cdiv((long long)N * HID, TB), TB, 0, stream>>>((unsigned*)O1, 0u, (long long)N * HID);

  gat_edge_max<<<cdiv((long long)nE * H1h, TB), TB, 0, stream>>>(ei, E, nE, as1, ad1, m1, H1h);
  gat_edge_sum<<<cdiv((long long)nE * H1h, TB), TB, 0, stream>>>(ei, E, nE, as1, ad1, m1, al1, s1, H1h);
  gat_edge_agg<<<nE, H1h * C1, 0, stream>>>(ei, E, H1, al1, s1, O1, H1h, C1);

  gat_bias_relu<<<cdiv((long long)N * HID, TB), TB, 0, stream>>>(O1, b1, HID, (long long)N * HID);

  // ================= Layer 2 =================
  {
    dim3 g(N / 16, C2 / 16);
    gat_gemm_wmma_f32<<<g, 32, 0, stream>>>(O1, W2, H2, N, C2, HID);
  }
  gat_att_scores<<<cdiv(N, TB), TB, 0, stream>>>(H2, aS2, aD2, as2, ad2, N, 1, C2);

  gat_fill_u32<<<cdiv(N, TB), TB, 0, stream>>>(m2, 0u, N);
  gat_fill_u32<<<cdiv(N, TB), TB, 0, stream>>>((unsigned*)s2, 0u, N);
  gat_init_bias<<<cdiv((long long)N * C2, TB), TB, 0, stream>>>(out, b2, C2, (long long)N * C2);

  gat_edge_max<<<cdiv(nE, TB), TB, 0, stream>>>(ei, E, nE, as2, ad2, m2, 1);
  gat_edge_sum<<<cdiv(nE, TB), TB, 0, stream>>>(ei, E, nE, as2, ad2, m2, al2, s2, 1);
  gat_edge_agg<<<nE, C2, 0, stream>>>(ei, E, H2, al2, s2, out, 1, C2);
}